// loTeNet_3D_68831145886561
// MI455X (gfx1250) — compile-verified
//
#include <hip/hip_runtime.h>
#include <math.h>
#include <stdint.h>

typedef __attribute__((ext_vector_type(2))) float v2f;
typedef __attribute__((ext_vector_type(8))) float v8f;
typedef __attribute__((ext_vector_type(4))) unsigned int v4u;
typedef __attribute__((ext_vector_type(4))) int v4i;
typedef __attribute__((ext_vector_type(8))) int v8i;

#define DBOND 32
#define NSITES 27
#define NMID 25
#define SITE_STRIDE   (DBOND * 2 * DBOND)        // 2048 floats per mid site (8KB)
#define CH_MID_STRIDE (NMID * SITE_STRIDE)       // 51200 floats per chain

// ---- Tensor Data Mover: stage one 8KB mid-site block global -> LDS ----
// D# per CDNA5 ISA: 1-D tile of 2048 x 4B elements.
__device__ __forceinline__ void tdm_load_site(const float* gsrc, uint32_t lds_off) {
  uint64_t ga = (uint64_t)(uintptr_t)gsrc;
  v4u g0;
  g0.x = 1u;                                              // count=1, user D#
  g0.y = lds_off;                                         // lds_addr (bytes)
  g0.z = (uint32_t)ga;                                    // global_addr[31:0]
  g0.w = (uint32_t)((ga >> 32) & 0x01FFFFFFu) | (2u << 30); // addr[56:32] | type=2
  v8i g1;
  g1[0] = (int)(2u << 16);       // workgroup_mask=0 | data_size=2 (4 bytes)
  g1[1] = (int)(2048u << 16);    // tensor_dim0[15:0]=2048 in bits[63:48]
  g1[2] = (int)(1u << 16);       // tensor_dim0[31:16]=0 | tensor_dim1=1
  g1[3] = (int)(2048u << 16);    // tensor_dim1[31:16]=0 | tile_dim0=2048
  g1[4] = 0;                     // tile_dim1=0 (1-D tile), tile_dim2=0
  g1[5] = 2048;                  // tensor_dim0_stride[31:0]
  g1[6] = 0;
  g1[7] = 0;
  v4i z4 = {0, 0, 0, 0};
#if __clang_major__ >= 23
  v8i z8 = {0, 0, 0, 0, 0, 0, 0, 0};
  __builtin_amdgcn_tensor_load_to_lds(g0, g1, z4, z4, z8, 0);
#else
  __builtin_amdgcn_tensor_load_to_lds(g0, g1, z4, z4, 0);
#endif
}

// ---- one chain-product step: P <- P @ (c*M0 + s*M1), all via WMMA f32 16x16x4 ----
__device__ __forceinline__ void chain_step(float* __restrict__ P,
                                           const float* __restrict__ ms, // LDS staged site
                                           float c, float s, int L) {
  const int lrow  = L & 15;
  const int khalf = (L >> 4) * 2;

  // Load whole P from LDS as WMMA A-operands (16x4 f32 chunks).
  v2f A[2][8];
  #pragma unroll
  for (int ti = 0; ti < 2; ++ti) {
    #pragma unroll
    for (int kc = 0; kc < 8; ++kc) {
      int r  = ti * 16 + lrow;
      int cc = kc * 4 + khalf;
      A[ti][kc] = *(const v2f*)&P[r * DBOND + cc];
    }
  }

  v8f Cz;
  #pragma unroll
  for (int i = 0; i < 8; ++i) Cz[i] = 0.0f;
  v8f C[2][2] = {{Cz, Cz}, {Cz, Cz}};

  #pragma unroll
  for (int kc = 0; kc < 8; ++kc) {
    #pragma unroll
    for (int tj = 0; tj < 2; ++tj) {
      // B chunk (kc,tj): 4x16 slice of mats, cos/sin combine fused on LDS reads.
      int colb = tj * 16 + lrow;
      int k0   = kc * 4 + khalf;
      v2f B;
      B.x = fmaf(c, ms[k0 * 64 + colb],       s * ms[k0 * 64 + 32 + colb]);
      B.y = fmaf(c, ms[(k0 + 1) * 64 + colb], s * ms[(k0 + 1) * 64 + 32 + colb]);
      #pragma unroll
      for (int ti = 0; ti < 2; ++ti) {
        C[ti][tj] = __builtin_amdgcn_wmma_f32_16x16x4_f32(
            false, A[ti][kc], false, B, (short)0, C[ti][tj], false, false);
      }
    }
  }

  // Scatter D tiles back to row-major P in LDS.
  #pragma unroll
  for (int ti = 0; ti < 2; ++ti) {
    #pragma unroll
    for (int tj = 0; tj < 2; ++tj) {
      #pragma unroll
      for (int r = 0; r < 8; ++r) {
        int M = ti * 16 + r + ((L >> 4) * 8);
        int N = tj * 16 + lrow;
        P[M * DBOND + N] = C[ti][tj][r];
      }
    }
  }
}

// One wave (32 lanes) per MPS chain. mode: 0 = layer0 (27^3 image input),
// 1 = layer1 (9^3 prev outputs), 2 = final (27 prev outputs, O=10).
__launch_bounds__(32)
__global__ void mps_layer_kernel(const float* __restrict__ in_vals,
                                 const float* __restrict__ first,
                                 const float* __restrict__ mid,
                                 const float* __restrict__ last,
                                 float* __restrict__ out,
                                 int mode)
{
  __shared__ __align__(16) float P[DBOND * DBOND];      // running 32x32 chain product
  __shared__ __align__(16) float smid[2][SITE_STRIDE];  // double-buffered TDM stage
  __shared__ float cosv[NSITES];
  __shared__ float sinv[NSITES];
  __shared__ float v0buf[10 * DBOND];

  const int n = blockIdx.x;
  const int L = threadIdx.x;

  // ---- gather this chain's 27 site values (squeeze_3d) and trig-embed ----
  if (L < NSITES) {
    int x = L / 9, y = (L / 3) % 3, z = L % 3;  // feature = x*9 + y*3 + z
    float val;
    if (mode == 0) {
      int bh = n / 81, bv = (n / 9) % 9, bd = n % 9;
      val = in_vals[(((3 * bh + x) * 27) + (3 * bv + y)) * 27 + (3 * bd + z)];
    } else if (mode == 1) {
      int bh = n / 9, bv = (n / 3) % 3, bd = n % 3;
      val = in_vals[(3 * bh + x) * 81 + (3 * bv + y) * 9 + (3 * bd + z)];
    } else {
      val = in_vals[L];
    }
    float a = 1.5707963267948966f * val;
    cosv[L] = cosf(a);
    sinv[L] = sinf(a);
  }
  __syncthreads();

  const float*  cmid = mid + (size_t)n * CH_MID_STRIDE;
  const uint32_t lds0 = (uint32_t)(uintptr_t)&smid[0][0];
  const uint32_t lds1 = (uint32_t)(uintptr_t)&smid[1][0];

  // ---- prologue: stage sites 0 and 1 via TDM ----
  tdm_load_site(cmid, lds0);
  tdm_load_site(cmid + SITE_STRIDE, lds1);
  __builtin_amdgcn_s_wait_tensorcnt((short)1);  // site 0 resident

  // ---- P = mats(site 1) = cos*M0 + sin*M1, row-major in LDS ----
  {
    float c = cosv[1], s = sinv[1];
    #pragma unroll 4
    for (int d = 0; d < DBOND; ++d)
      P[d * DBOND + L] = fmaf(c, smid[0][d * 64 + L], s * smid[0][d * 64 + 32 + L]);
  }
  __syncthreads();

  // ---- steady state: issue TDM for site m+1, wait site m, multiply it in ----
  for (int m = 1; m < NMID - 1; ++m) {
    tdm_load_site(cmid + (m + 1) * SITE_STRIDE, ((m + 1) & 1) ? lds1 : lds0);
    __builtin_amdgcn_s_wait_tensorcnt((short)1);   // oldest copy (site m) done
    chain_step(P, smid[m & 1], cosv[m + 1], sinv[m + 1], L);
    __syncthreads();
  }
  // ---- last mid site ----
  __builtin_amdgcn_s_wait_tensorcnt((short)0);
  chain_step(P, smid[(NMID - 1) & 1], cosv[NMID], sinv[NMID], L);
  __syncthreads();

  // ---- final contraction: out[o] = sum_{d,e} v0[o,d] * P[d,e] * w[e] ----
  const int O = (mode == 2) ? 10 : 1;
  {
    float c0 = cosv[0], s0 = sinv[0];
    const float* f = (mode == 2) ? first : (first + (size_t)n * 2 * DBOND);
    for (int o = 0; o < O; ++o) {
      // first layout: [p][o][d]; p-stride = O*DBOND
      v0buf[o * DBOND + L] =
          fmaf(c0, f[o * DBOND + L], s0 * f[O * DBOND + o * DBOND + L]);
    }
  }
  __syncthreads();

  const float* lp = (mode == 2) ? last : (last + (size_t)n * DBOND * 2);
  float w = fmaf(cosv[NSITES - 1], lp[L * 2], sinv[NSITES - 1] * lp[L * 2 + 1]);

  for (int o = 0; o < O; ++o) {
    float acc = 0.0f;
    #pragma unroll 8
    for (int d = 0; d < DBOND; ++d)
      acc = fmaf(v0buf[o * DBOND + d], P[d * DBOND + L], acc);
    acc *= w;
    #pragma unroll
    for (int off = 16; off > 0; off >>= 1)
      acc += __shfl_xor(acc, off, 32);
    if (L == 0) out[(mode == 2) ? o : n] = acc;
  }
}

extern "C" void kernel_launch(void* const* d_in, const int* in_sizes, int n_in,
                              void* d_out, int out_size, void* d_ws, size_t ws_size,
                              hipStream_t stream) {
  (void)in_sizes; (void)n_in; (void)out_size; (void)ws_size;
  const float* img      = (const float*)d_in[0];
  const float* l0_first = (const float*)d_in[1];
  const float* l0_mid   = (const float*)d_in[2];
  const float* l0_last  = (const float*)d_in[3];
  const float* l1_first = (const float*)d_in[4];
  const float* l1_mid   = (const float*)d_in[5];
  const float* l1_last  = (const float*)d_in[6];
  const float* f_first  = (const float*)d_in[7];
  const float* f_mid    = (const float*)d_in[8];
  const float* f_last   = (const float*)d_in[9];

  float* out0 = (float*)d_ws;     // 729 layer-0 scalars
  float* out1 = out0 + 729;       // 27 layer-1 scalars
  float* outp = (float*)d_out;    // 10 logits

  mps_layer_kernel<<<729, 32, 0, stream>>>(img,  l0_first, l0_mid, l0_last, out0, 0);
  mps_layer_kernel<<<27,  32, 0, stream>>>(out0, l1_first, l1_mid, l1_last, out1, 1);
  mps_layer_kernel<<<1,   32, 0, stream>>>(out1, f_first,  f_mid,  f_last,  outp, 2);
}